// Kernel_23536420782426
// MI455X (gfx1250) — compile-verified
//
#include <hip/hip_runtime.h>

// V_WMMA_F32_16X16X4_F32 operand types (wave32):
//  A: 16x4 f32  -> 2 VGPRs/lane   (lane l: M=l%16; VGPR v: K = v + 2*(l/16))
//  B:  4x16 f32 -> 2 VGPRs/lane   (lane l: N=l%16; VGPR v: K = v + 2*(l/16))
//  C/D: 16x16 f32 -> 8 VGPRs/lane (lane l: N=l%16; VGPR v: M = v + 8*(l/16))
typedef __attribute__((ext_vector_type(2))) float v2f;
typedef __attribute__((ext_vector_type(8))) float v8f;

#define IMG_W   512
#define DEPTH   64
#define OUT_N   510
#define NOUT    (OUT_N * OUT_N)          // 260100 outputs
#define NTILES  ((NOUT + 15) / 16)       // 16257 wave-tiles of 16 pixels
#define KTOT    (3 * 3 * DEPTH)          // 576-long dot product

__launch_bounds__(256)
__global__ void conv3x3d64_wmma_f32(const float* __restrict__ x,
                                    const float* __restrict__ W,
                                    const float* __restrict__ bias,
                                    float* __restrict__ out)
{
    // Stage the 576 weights in LDS once per workgroup (2.3 KB).
    __shared__ float wlds[KTOT];
    for (int i = threadIdx.x; i < KTOT; i += 256) wlds[i] = W[i];
    __syncthreads();

    const int wave = threadIdx.x >> 5;          // 8 waves / block
    const int lane = threadIdx.x & 31;
    const int tile = blockIdx.x * 8 + wave;
    if (tile >= NTILES) return;                 // wave-uniform: EXEC stays all-1s

    const int      n  = lane & 15;              // B-matrix column = pixel slot
    const int      h  = lane >> 4;              // K-subrow pair selector (0/1)
    const unsigned p  = (unsigned)tile * 16u + (unsigned)n;
    const unsigned pe = (p < NOUT) ? p : (NOUT - 1);   // clamp, keep EXEC full
    const unsigned r  = pe / OUT_N;
    const unsigned c  = pe % OUT_N;

    // Per-lane base pointers; all loop offsets below are compile-time
    // immediates folded into the global_load/ds_load instruction offsets.
    const float* xb = x    + ((size_t)r * IMG_W + c) * DEPTH + 2 * h;
    const float* wb = wlds + 2 * h;

    v8f acc = {};
#pragma unroll
    for (int fi = 0; fi < 3; ++fi) {
#pragma unroll
        for (int fj = 0; fj < 3; ++fj) {
#pragma unroll
            for (int d0 = 0; d0 < DEPTH; d0 += 4) {
                const int xoff = fi * (IMG_W * DEPTH) + fj * DEPTH + d0;
                const int koff = (fi * 3 + fj) * DEPTH + d0;
                // B: patch data, one pixel per column. 8B-aligned b64 load.
                v2f bvec = *(const v2f*)(xb + xoff);
                // A: weights replicated across all 16 M-rows (ds_load_b64).
                v2f avec = *(const v2f*)(wb + koff);
                // D = A*B + C : every D row holds the 16 pixel results.
                acc = __builtin_amdgcn_wmma_f32_16x16x4_f32(
                        /*neg_a=*/false, avec, /*neg_b=*/false, bvec,
                        /*c_mod=*/(short)0, acc,
                        /*reuse_a=*/false, /*reuse_b=*/false);
            }
        }
    }

    // Lane l (l<16) owns pixel p in column n=l; row M=0 lives in acc[0].
    if (h == 0 && p < NOUT) out[p] = acc[0] + bias[0];
}

extern "C" void kernel_launch(void* const* d_in, const int* in_sizes, int n_in,
                              void* d_out, int out_size, void* d_ws, size_t ws_size,
                              hipStream_t stream) {
    (void)in_sizes; (void)n_in; (void)d_ws; (void)ws_size; (void)out_size;
    const float* x  = (const float*)d_in[0];
    const float* W  = (const float*)d_in[1];
    const float* b  = (const float*)d_in[2];
    float* out      = (float*)d_out;

    const int blocks = (NTILES + 7) / 8;   // 8 wave-tiles per 256-thread block
    hipLaunchKernelGGL(conv3x3d64_wmma_f32, dim3(blocks), dim3(256), 0, stream,
                       x, W, b, out);
}